// TotalLoss_38671885533270
// MI455X (gfx1250) — compile-verified
//
#include <hip/hip_runtime.h>
#include <hip/hip_bf16.h>

#define EPSF 1.1920928955078125e-07f   // float32 machine epsilon

typedef __attribute__((ext_vector_type(2))) float v2f;
typedef __attribute__((ext_vector_type(8))) float v8f;

// Full-precision 32-lane reduction via V_WMMA_F32_16X16X4_F32.
// A is 16x4: lane l<16 holds A[l,0..1], lane l>=16 holds A[l-16,2..3].
// With a = {v, 0} and B = ones(4x16):  D[m,n] = v_m + v_{m+16}.
// Lane n (<16) holds D[0..7,n] in its 8 dest VGPRs, lane n+16 holds D[8..15,n],
// so sum-of-8 + one xor-16 shuffle yields the full wave sum in every lane.
// Must be called with EXEC all-ones (no divergence at call sites).
__device__ __forceinline__ float wave_reduce_f32(float v) {
    v2f a; a.x = v;    a.y = 0.0f;
    v2f b; b.x = 1.0f; b.y = 1.0f;
    v8f c = {0.f, 0.f, 0.f, 0.f, 0.f, 0.f, 0.f, 0.f};
    v8f d = __builtin_amdgcn_wmma_f32_16x16x4_f32(
        /*neg_a=*/false, a, /*neg_b=*/false, b,
        /*c_mod=*/(short)0, c, /*reuse_a=*/false, /*reuse_b=*/false);
    float s = d[0] + d[1] + d[2] + d[3] + d[4] + d[5] + d[6] + d[7];
    s += __shfl_xor(s, 16, 32);
    return s;
}

__global__ void zero_f32_kernel(float* p, int n) {
    int stride = gridDim.x * blockDim.x;
    for (int i = blockIdx.x * blockDim.x + threadIdx.x; i < n; i += stride)
        p[i] = 0.0f;
}

// ---- weight decay: sum(params^2) -------------------------------------------
__global__ void wd_kernel(const float* __restrict__ params, float* acc, int n) {
    float local = 0.0f;
    int stride = gridDim.x * blockDim.x;
    for (int i = blockIdx.x * blockDim.x + threadIdx.x; i < n; i += stride) {
        float x = params[i];
        local += x * x;
    }
    float s = wave_reduce_f32(local);
    if ((threadIdx.x & 31) == 0) atomicAdd(acc, s);
}

// ---- smoothness: 4 charbonnier terms on one flow (B,2,H,W) -----------------
__global__ void smooth_kernel(const float* __restrict__ f, float* acc,
                              int B, int H, int W) {
    const float e2 = 1.0e-6f;  // epsilon^2, epsilon = 0.001
    const long total = (long)B * 2 * H * W;
    const float inv1 = 1.0f / ((float)B * 2.0f * (float)H * (float)(W - 1));
    const float inv2 = 1.0f / ((float)B * 2.0f * (float)(H - 1) * (float)W);
    const float inv3 = 1.0f / ((float)B * 2.0f * (float)(H - 1) * (float)(W - 1));
    float local = 0.0f;
    long stride = (long)gridDim.x * blockDim.x;
    for (long i = blockIdx.x * (long)blockDim.x + threadIdx.x; i < total; i += stride) {
        int  x  = (int)(i % W);
        long r  = i / W;
        int  y  = (int)(r % H);
        long bc = r / H;                       // in [0, 2B)
        const float* base = f + (bc * H + y) * W;
        float c = base[x];
        if (x + 1 < W) {
            float dd = base[x + 1] - c;
            local += powf(dd * dd + e2, 0.45f) * inv1;
        }
        if (y + 1 < H) {
            float dd = base[W + x] - c;
            local += powf(dd * dd + e2, 0.45f) * inv2;
        }
        if (x + 1 < W && y + 1 < H) {
            float d3 = base[W + x + 1] - c;            // ul - dr
            float d4 = base[x + 1] - base[W + x];      // dl - ur
            local += (powf(d3 * d3 + e2, 0.45f) + powf(d4 * d4 + e2, 0.45f)) * inv3;
        }
    }
    local *= 0.25f;  // compute_smoothness_loss returns s/4
    float s = wave_reduce_f32(local);
    if ((threadIdx.x & 31) == 0) atomicAdd(acc, s);
}

// ---- event-loss scatter: bilinear atomics into den/num images --------------
// Accumulator layout (per flow): combo = (b*2 + dir)*2 + pol, combo in [0,16);
// den at acc[(combo*2+0)*WH], num at acc[(combo*2+1)*WH].
__global__ void scatter_kernel(const float* __restrict__ xs,
                               const float* __restrict__ ys,
                               const float* __restrict__ ts,
                               const int*   __restrict__ ps,
                               const float* __restrict__ flow,
                               float* __restrict__ acc,
                               int N, int W, int H, float invdiv, int flow_idx) {
    const long total = 4L * (long)N;
    const long WH    = (long)W * H;
    long stride = (long)gridDim.x * blockDim.x;
    for (long gid = blockIdx.x * (long)blockDim.x + threadIdx.x; gid < total; gid += stride) {
        int b = (int)(gid / N);
        // compose the (flow_idx+1) affine t-normalizations from two scalars
        float F = ts[(long)b * N];
        float L = ts[(long)b * N + (N - 1)];
        float alpha = 1.0f, beta = 0.0f;
        for (int k = 0; k <= flow_idx; ++k) {
            float Fk  = alpha * F + beta;
            float Lk  = alpha * L + beta;
            float inv = 1.0f / (Lk - Fk + EPSF);
            alpha = alpha * inv;
            beta  = (beta - Fk) * inv;
        }
        float tn = alpha * ts[gid] + beta;
        float Ln = alpha * L + beta;     // t_cur[:, -1]
        float Fn = alpha * F + beta;     // t_cur[:,  0]

        float xv = xs[gid] * invdiv;
        float yv = ys[gid] * invdiv;
        int xi = (int)xv;                // truncation, values >= 0
        int yi = (int)yv;
        int pol = (ps[gid] == 1) ? 0 : 1;

        float fx = flow[(((long)b * 2 + 0) * H + yi) * W + xi];
        float fy = flow[(((long)b * 2 + 1) * H + yi) * W + xi];

        for (int dir = 0; dir < 2; ++dir) {
            float tt = (dir == 0) ? (Ln - tn) : (Fn - tn);
            float x_ = fminf(fmaxf((float)xi * 255.0f + tt * fx, 0.0f), (float)(W - 1));
            float y_ = fminf(fmaxf((float)yi * 255.0f + tt * fy, 0.0f), (float)(H - 1));
            float x0 = floorf(x_), y0 = floorf(y_);
            float x1 = x0 + 1.0f,  y1 = y0 + 1.0f;
            float x0_ = x_ - x0, x1_ = x1 - x_;
            float y0_ = y_ - y0, y1_ = y1 - y_;
            float Ra = x0_ * y0_, Rb = x1_ * y0_, Rc = x0_ * y1_, Rd = x1_ * y1_;
            float Ta = Ra * tt,   Tb = Rb * tt,   Tc = Rc * tt,   Td = Rd * tt;
            int x1i = (int)fminf(x1, (float)(W - 1));
            int y1i = (int)fminf(y1, (float)(H - 1));
            int x0i = (int)x0, y0i = (int)y0;
            int Ia = x1i + y1i * W, Ib = x0i + y1i * W;
            int Ic = x1i + y0i * W, Id = x0i + y0i * W;
            long combo = ((long)(b * 2 + dir) * 2 + pol);
            float* den = acc + (combo * 2 + 0) * WH;
            float* num = acc + (combo * 2 + 1) * WH;
            atomicAdd(den + Ia, Ra + EPSF);  atomicAdd(num + Ia, Ta + EPSF);
            atomicAdd(den + Ib, Rb + EPSF);  atomicAdd(num + Ib, Tb + EPSF);
            atomicAdd(den + Ic, Rc + EPSF);  atomicAdd(num + Ic, Tc + EPSF);
            atomicAdd(den + Id, Rd + EPSF);  atomicAdd(num + Id, Td + EPSF);
        }
    }
}

// ---- event-loss reduce: sum((num/(den+EPS))^2) over 16 combos --------------
__global__ void ev_reduce_kernel(const float* __restrict__ acc, float* out_acc, long WH) {
    float local = 0.0f;
    const long total = 16L * WH;
    long stride = (long)gridDim.x * blockDim.x;
    for (long i = blockIdx.x * (long)blockDim.x + threadIdx.x; i < total; i += stride) {
        long combo = i / WH;
        long pix   = i % WH;
        float den = acc[(combo * 2 + 0) * WH + pix];
        float num = acc[(combo * 2 + 1) * WH + pix];
        float r = num / (den + EPSF);
        local += r * r;
    }
    float s = wave_reduce_f32(local);
    if ((threadIdx.x & 31) == 0) atomicAdd(out_acc, s);
}

__global__ void finalize_kernel(const float* __restrict__ ws, float* out) {
    // ws[0]=sum(p^2), ws[1]=sum_f smooth_f, ws[2]=raw event sum
    out[0] = ws[0] * (0.5f * 1.0e-4f)   // weight decay
           + ws[1] * 0.125f             // SMOOTHNESS_WEIGHT/4 = 0.5/4
           + ws[2] * (1.0f / 64.0f);    // 4 * len(flows) * B = 64
}

extern "C" void kernel_launch(void* const* d_in, const int* in_sizes, int n_in,
                              void* d_out, int out_size, void* d_ws, size_t ws_size,
                              hipStream_t stream) {
    const float* flows[4] = { (const float*)d_in[0], (const float*)d_in[1],
                              (const float*)d_in[2], (const float*)d_in[3] };
    const float* xs = (const float*)d_in[4];
    const float* ys = (const float*)d_in[5];
    const float* ts = (const float*)d_in[6];
    const int*   ps = (const int*)d_in[7];
    const float* params = (const float*)d_in[8];
    float* out = (float*)d_out;
    float* ws  = (float*)d_ws;
    float* acc = ws + 16;                      // 8 MB accumulator region, reused per flow

    const int B = 4;
    const int N = in_sizes[4] / B;             // 1,000,000
    const int nparams = in_sizes[8];           // 2048*1024

    // zero scalar accumulators
    zero_f32_kernel<<<1, 64, 0, stream>>>(ws, 16);

    // weight decay
    wd_kernel<<<512, 256, 0, stream>>>(params, ws + 0, nparams);

    const int   dims[4]   = { 32, 64, 128, 256 };
    const float invdiv[4] = { 1.0f / 8.0f, 1.0f / 32.0f, 1.0f / 64.0f, 1.0f / 64.0f };

    // smoothness over all 4 flows
    for (int f = 0; f < 4; ++f)
        smooth_kernel<<<256, 256, 0, stream>>>(flows[f], ws + 1, B, dims[f], dims[f]);

    // event loss, one flow at a time (accumulators are L2-resident, <= 8 MB)
    for (int f = 0; f < 4; ++f) {
        int W = dims[f], H = dims[f];
        long WH = (long)W * H;
        int nacc = 32 * W * H;
        int zb = (nacc + 255) / 256; if (zb > 4096) zb = 4096;
        zero_f32_kernel<<<zb, 256, 0, stream>>>(acc, nacc);

        long totalev = 4L * N;
        int sb = (int)((totalev + 255) / 256);           // 15625 blocks of 8 waves
        scatter_kernel<<<sb, 256, 0, stream>>>(xs, ys, ts, ps, flows[f], acc,
                                               N, W, H, invdiv[f], f);

        ev_reduce_kernel<<<1024, 256, 0, stream>>>(acc, ws + 2, WH);
    }

    finalize_kernel<<<1, 1, 0, stream>>>(ws, out);
}